// Mind3D_85976655331978
// MI455X (gfx1250) — compile-verified
//
#include <hip/hip_runtime.h>
#include <stdint.h>

// ---------------------------------------------------------------------------
// MIND-SSC 3D pipeline for MI455X (gfx1250, wave32).
// Memory-bound: ~1.9 GB total HBM traffic across 6 streaming passes
// (~80 us at 23.3 TB/s). Separable 5^3 box filter:
//   box-x: TDM (tensor_load_to_lds) stages rows through LDS
//   box-y: stencil-as-GEMM on the matrix pipe (v_wmma_f32_16x16x4_f32,
//          fixed 16x20 band matrix, clamped B-rows handle edge replication)
//   box-z: fused with channel-min/mind + global-mean double accumulation
// Workspace: 256 B scalars + one 211 MB ping-pong buffer (d_out is the other).
// ---------------------------------------------------------------------------

#define ON   130                    // output spatial extent
#define DN   128                    // input spatial extent
#define BN   2
#define CN   12
static constexpr int SPC  = ON * ON * ON;     // 2,197,000 spatial per channel
static constexpr int NVOX = BN * SPC;         // 4,394,000
static constexpr int NTOT = BN * CN * SPC;    // 52,728,000
static constexpr int NX   = BN * DN * DN * DN;

// 2*P shift points (z,y,x) and the 12 (i>j, d2==2) pairs, hardcoded from the
// reference's _mind_pairs() construction.
__device__ __constant__ int SZ6[6] = {0, 2, 2, 2, 4, 2};
__device__ __constant__ int SY6[6] = {2, 2, 0, 2, 2, 4};
__device__ __constant__ int SX6[6] = {2, 0, 2, 4, 2, 2};
__device__ __constant__ int PA[12] = {1, 2, 2, 3, 3, 4, 4, 4, 5, 5, 5, 5};
__device__ __constant__ int PB[12] = {0, 0, 1, 0, 2, 1, 2, 3, 0, 1, 3, 4};

__device__ __forceinline__ int iclamp(int v, int lo, int hi) {
    return v < lo ? lo : (v > hi ? hi : v);
}
// Order-preserving float<->uint keys (works for any sign).
__device__ __forceinline__ unsigned fenc(float f) {
    unsigned u = __float_as_uint(f);
    return (u & 0x80000000u) ? ~u : (u | 0x80000000u);
}
__device__ __forceinline__ float fdec(unsigned e) {
    unsigned u = (e & 0x80000000u) ? (e ^ 0x80000000u) : ~e;
    return __uint_as_float(u);
}

// ---------------------------------------------------------------- K0: init --
__global__ void k_init(unsigned* mnmx, double* accum) {
    if (threadIdx.x == 0) {
        mnmx[0] = 0xFFFFFFFFu;  // running min key
        mnmx[1] = 0u;           // running max key
        *accum  = 0.0;
    }
}

// ------------------------------------------------------------- K1: min/max --
__global__ __launch_bounds__(256) void k_minmax(const float* __restrict__ x,
                                                unsigned* mnmx) {
    unsigned lmn = 0xFFFFFFFFu, lmx = 0u;
    for (int i = blockIdx.x * blockDim.x + threadIdx.x; i < NX;
         i += gridDim.x * blockDim.x) {
        unsigned k = fenc(x[i]);
        lmn = min(lmn, k);
        lmx = max(lmx, k);
    }
    __shared__ unsigned rmn[256], rmx[256];
    rmn[threadIdx.x] = lmn; rmx[threadIdx.x] = lmx;
    __syncthreads();
    for (int st = 128; st > 0; st >>= 1) {
        if ((int)threadIdx.x < st) {
            rmn[threadIdx.x] = min(rmn[threadIdx.x], rmn[threadIdx.x + st]);
            rmx[threadIdx.x] = max(rmx[threadIdx.x], rmx[threadIdx.x + st]);
        }
        __syncthreads();
    }
    if (threadIdx.x == 0) {
        atomicMin(&mnmx[0], rmn[0]);
        atomicMax(&mnmx[1], rmx[0]);
    }
}

// -------------------------------------- K2: normalize + 12 shifted diffs^2 --
// xp(p) (padded coord in [0,134)): zero if p==0||p==133, else
// normalized x[clamp(p-3,0,127)].  Six distinct sample points per voxel.
__global__ __launch_bounds__(256) void k_diff(const float* __restrict__ x,
                                              float* __restrict__ out,
                                              const unsigned* __restrict__ mnmx) {
    int t = blockIdx.x * blockDim.x + threadIdx.x;
    if (t >= NVOX) return;
    int b = t / SPC;
    int s = t - b * SPC;
    int z = s / (ON * ON);
    int r = s - z * ON * ON;
    int y = r / ON;
    int xo = r - y * ON;

    float vmin = fdec(mnmx[0]);
    float vmax = fdec(mnmx[1]);
    float inv  = 1.0f / (vmax - vmin + 1e-6f);

    const float* xb = x + b * (DN * DN * DN);
    float smp[6];
#pragma unroll
    for (int k = 0; k < 6; ++k) {
        int pz = z + SZ6[k], py = y + SY6[k], px = xo + SX6[k];
        bool zb = (pz == 0) | (pz == 133) | (py == 0) | (py == 133) |
                  (px == 0) | (px == 133);
        int iz = iclamp(pz - 3, 0, 127);
        int iy = iclamp(py - 3, 0, 127);
        int ix = iclamp(px - 3, 0, 127);
        float v = xb[(iz * DN + iy) * DN + ix];
        smp[k] = zb ? 0.0f : (v - vmin) * inv;
    }
    float* ob = out + b * (CN * SPC) + s;
#pragma unroll
    for (int c = 0; c < 12; ++c) {
        float d = smp[PA[c]] - smp[PB[c]];
        ob[c * SPC] = d * d;
    }
}

// ------------------------------------------------- K3: box-x via TDM + LDS --
// One wave handles 4 contiguous x-rows (520 floats) staged through LDS by the
// Tensor Data Mover; edge handling = clamped index within the 130-float row.
#define K3_WAVES 8
#define K3_CHUNK 520   /* 4 rows * 130 */

typedef unsigned int v4u __attribute__((ext_vector_type(4)));
typedef int          v8i __attribute__((ext_vector_type(8)));
typedef int          v4i __attribute__((ext_vector_type(4)));
typedef float        v2f __attribute__((ext_vector_type(2)));
typedef float        v8f __attribute__((ext_vector_type(8)));

#if defined(__has_builtin)
#if __has_builtin(__builtin_amdgcn_tensor_load_to_lds) && \
    __has_builtin(__builtin_amdgcn_s_wait_tensorcnt)
#define USE_TDM 1
#endif
#endif
#ifndef USE_TDM
#define USE_TDM 0
#endif

__global__ __launch_bounds__(256) void k_boxx(const float* __restrict__ in,
                                              float* __restrict__ out) {
    __shared__ float smem[K3_WAVES * K3_CHUNK];
    int wid  = __builtin_amdgcn_readfirstlane((int)(threadIdx.x >> 5));
    int lane = (int)(threadIdx.x & 31);
    int chunk = blockIdx.x * K3_WAVES + wid;        // [0, 101400)
    int base  = chunk * K3_CHUNK;                   // element offset (< 2^31)
    float* my = smem + wid * K3_CHUNK;

#if USE_TDM
    // Build Tensor DMA descriptor (D#) per CDNA5 ISA.
    uint64_t ga = (uint64_t)(uintptr_t)(const void*)(in + base);
    unsigned lds = (unsigned)(uintptr_t)(void*)my;  // flat low32 == LDS offset
    v4u g0;
    g0.x = 1u;                                      // count=1 (valid), user D#
    g0.y = lds;                                     // lds_addr
    g0.z = (unsigned)ga;                            // global_addr[31:0]
    g0.w = (unsigned)((ga >> 32) & 0x1FFFFFFu) | (2u << 30);  // addr hi, type=2
    v8i g1;
    g1[0] = (int)(2u << 16);                        // data_size=4B, mask=0
    g1[1] = (int)((unsigned)K3_CHUNK << 16);        // tensor_dim0[15:0]
    g1[2] = (int)(1u << 16);                        // tensor_dim0 hi=0, dim1=1
    g1[3] = (int)((unsigned)K3_CHUNK << 16);        // dim1 hi=0, tile_dim0=520
    g1[4] = 1;                                      // tile_dim1=1, tile_dim2=0
    g1[5] = K3_CHUNK;                               // tensor_dim0_stride
    g1[6] = 0;
    g1[7] = 0;
    v4i zg4 = {0, 0, 0, 0};
    v8i zg8 = {0, 0, 0, 0, 0, 0, 0, 0};
    __builtin_amdgcn_tensor_load_to_lds(g0, g1, zg4, zg4, zg8, 0);
    __builtin_amdgcn_s_wait_tensorcnt(0);
    asm volatile("" ::: "memory");                  // LDS now holds the tile
#else
    for (int e = lane; e < K3_CHUNK; e += 32) my[e] = in[base + e];
    __syncthreads();
#endif

    for (int e = lane; e < K3_CHUNK; e += 32) {
        int rr = e / 130;
        int xo = e - rr * 130;
        const float* row = my + rr * 130;
        float sum = row[max(xo - 2, 0)] + row[max(xo - 1, 0)] + row[xo] +
                    row[min(xo + 1, 129)] + row[min(xo + 2, 129)];
        out[base + e] = sum;
    }
}

// ------------------------------------------- K4: box-y as WMMA band-GEMM ----
// out_tile(16y x 16x) = A(16x20 band of ones) * in(20 clamped rows x 16x),
// decomposed into 5 chained V_WMMA_F32_16X16X4_F32 ops (K chunks of 4).
// ISA layouts (CDNA5 ISA 7.12.2, 32-bit):
//   A 16x4 : M = lane&15, VGPR{0,1} = K{k0,k0+1}, k0 = 2*(lane>>4)
//   B 4x16 : N = lane&15, VGPR{0,1} = K{k0,k0+1}, k0 = 2*(lane>>4)
//   D 16x16: N = lane&15, VGPR r = row r + 8*(lane>>4)
// Edge replication: B-row index clamped (each tap -> distinct k). Partial
// x-tiles: clamp load column, guard stores. EXEC all-ones until stores.
#define YT 9            /* ceil(130/16) y tiles */
#define XT 9            /* ceil(130/16) x tiles */

__global__ __launch_bounds__(256) void k_boxy_wmma(const float* __restrict__ in,
                                                   float* __restrict__ out) {
    int wid  = __builtin_amdgcn_readfirstlane((int)(threadIdx.x >> 5));
    int lane = (int)(threadIdx.x & 31);
    int w  = blockIdx.x * 8 + wid;          // tile id, [0, 24*130*81)
    int xt = w % XT;
    int t1 = w / XT;
    int yt = t1 % YT;
    int t2 = t1 / YT;
    int z  = t2 % ON;
    int bc = t2 / ON;                       // [0, BN*CN)

    int y0 = yt * 16, x0 = xt * 16;
    int n    = lane & 15;
    int half = lane >> 4;
    int xc   = min(x0 + n, ON - 1);         // clamped load column
    const float* plane = in + bc * SPC + z * (ON * ON);

    v8f acc = {0.f, 0.f, 0.f, 0.f, 0.f, 0.f, 0.f, 0.f};
#pragma unroll
    for (int c5 = 0; c5 < 5; ++c5) {
        int k0 = 4 * c5 + 2 * half;         // this lane's K pair
        // Band matrix A[m][k] = (m <= k <= m+4), m = lane&15.
        v2f a;
        a.x = (k0 >= n && k0 <= n + 4) ? 1.0f : 0.0f;
        a.y = (k0 + 1 >= n && k0 + 1 <= n + 4) ? 1.0f : 0.0f;
        // B rows: input rows y0-2+k, edge-clamped.
        int r0 = iclamp(y0 - 2 + k0, 0, ON - 1);
        int r1 = iclamp(y0 - 1 + k0, 0, ON - 1);
        v2f b;
        b.x = plane[r0 * ON + xc];
        b.y = plane[r1 * ON + xc];
        acc = __builtin_amdgcn_wmma_f32_16x16x4_f32(
            /*neg_a=*/false, a, /*neg_b=*/false, b,
            /*c_mod=*/(short)0, acc, /*reuse_a=*/false, /*reuse_b=*/false);
    }

    float* oplane = out + bc * SPC + z * (ON * ON);
    int xs = x0 + n;
#pragma unroll
    for (int r = 0; r < 8; ++r) {
        int row = y0 + r + 8 * half;
        if (row < ON && xs < ON) oplane[row * ON + xs] = acc[r];
    }
}

// ---------------------- K5: box-z + channel-min + mind + global mean accum --
__global__ __launch_bounds__(256) void k_boxz_mind(const float* __restrict__ in,
                                                   float* __restrict__ out,
                                                   double* __restrict__ accum) {
    int t = blockIdx.x * blockDim.x + threadIdx.x;
    float var = 0.0f;
    if (t < NVOX) {
        int b = t / SPC;
        int s = t - b * SPC;
        int z = s / (ON * ON);
        int r = s - z * ON * ON;                    // y*ON + x
        const float* base = in + b * (CN * SPC) + r;
        float ssd[12];
#pragma unroll
        for (int c = 0; c < 12; ++c) {
            float sum = 0.0f;
#pragma unroll
            for (int d = -2; d <= 2; ++d) {
                int zz = iclamp(z + d, 0, ON - 1);
                sum += base[c * SPC + zz * (ON * ON)];
            }
            ssd[c] = sum * (1.0f / 125.0f);
        }
        float mn = ssd[0];
#pragma unroll
        for (int c = 1; c < 12; ++c) mn = fminf(mn, ssd[c]);
        float* ob = out + b * (CN * SPC) + s;
        float acc = 0.0f;
#pragma unroll
        for (int c = 0; c < 12; ++c) {
            float m = ssd[c] - mn;
            ob[c * SPC] = m;
            acc += m;
        }
        var = acc * (1.0f / 12.0f);
    }
    __shared__ float red[256];
    red[threadIdx.x] = var;
    __syncthreads();
    for (int st = 128; st > 0; st >>= 1) {
        if ((int)threadIdx.x < st) red[threadIdx.x] += red[threadIdx.x + st];
        __syncthreads();
    }
    if (threadIdx.x == 0) atomicAdd(accum, (double)red[0]);
}

// ------------------------------------------------- K6: clip + exp(-m/var) ---
__global__ __launch_bounds__(256) void k_exp(const float* __restrict__ in,
                                             float* __restrict__ out,
                                             const double* __restrict__ accum) {
    int t = blockIdx.x * blockDim.x + threadIdx.x;
    if (t >= NVOX) return;
    float m = (float)(*accum / (double)NVOX);       // global mean of mind_var
    int b = t / SPC;
    int s = t - b * SPC;
    const float* ib = in + b * (CN * SPC) + s;
    float* ob = out + b * (CN * SPC) + s;
    float v[12];
    float acc = 0.0f;
#pragma unroll
    for (int c = 0; c < 12; ++c) {
        v[c] = ib[c * SPC];
        acc += v[c];
    }
    float var = acc * (1.0f / 12.0f);
    var = fminf(fmaxf(var, m * 0.001f), m * 1000.0f);
    float ninv = -1.0f / var;
#pragma unroll
    for (int c = 0; c < 12; ++c) ob[c * SPC] = __expf(v[c] * ninv);
}

// ---------------------------------------------------------------------------
extern "C" void kernel_launch(void* const* d_in, const int* in_sizes, int n_in,
                              void* d_out, int out_size, void* d_ws, size_t ws_size,
                              hipStream_t stream) {
    (void)in_sizes; (void)n_in; (void)out_size; (void)ws_size;
    const float* x   = (const float*)d_in[0];
    float*       out = (float*)d_out;

    char*     ws    = (char*)d_ws;
    double*   accum = (double*)ws;                  // 8 B
    unsigned* mnmx  = (unsigned*)(ws + 8);          // 2 x u32
    float*    bufA  = (float*)(ws + 256);           // NTOT floats (~211 MB)

    k_init<<<1, 64, 0, stream>>>(mnmx, accum);
    k_minmax<<<1024, 256, 0, stream>>>(x, mnmx);
    k_diff<<<(NVOX + 255) / 256, 256, 0, stream>>>(x, out, mnmx);
    // box-x: d_out -> bufA   (101400 wave-chunks of 520, 8 waves/block)
    k_boxx<<<(NTOT / K3_CHUNK) / K3_WAVES, 256, 0, stream>>>(out, bufA);
    // box-y (WMMA band-GEMM): bufA -> d_out   (24*130*81 tiles, 8 waves/block)
    k_boxy_wmma<<<(BN * CN * ON * YT * XT) / 8, 256, 0, stream>>>(bufA, out);
    // box-z + mind + mean accum: d_out -> bufA
    k_boxz_mind<<<(NVOX + 255) / 256, 256, 0, stream>>>(out, bufA, accum);
    // final: bufA -> d_out
    k_exp<<<(NVOX + 255) / 256, 256, 0, stream>>>(bufA, out, accum);
}